// GraphMemory_53996328845716
// MI455X (gfx1250) — compile-verified
//
#include <hip/hip_runtime.h>
#include <hip/hip_bf16.h>

// Problem constants (B=4, D=48)
#define DD 48
#define ROW 2304           // 48*48, contiguous inner length of every big tensor
#define NROWS 9216         // 4 * 2304 rows of the (B,d,d,d,d) tensors
#define LRF 0.01f
#define INV_SQRT2 0.70710678118654752440f
#define INV_N (1.0f / 9216.0f)   // 1 / (B*D*D), huber mean denominator

typedef __attribute__((ext_vector_type(2))) float v2f;
typedef __attribute__((ext_vector_type(4))) float v4f;   // true clang vector -> b128 + NT hints
typedef __attribute__((ext_vector_type(8))) float v8f;

// ---------------------------------------------------------------------------
// Kernel 1 (fused, memory-bound path): per row (b,q,w) of length 2304:
//   - updated = space + bias + kernel * x[b,q,Q]/sqrt(2) at W==w
//   - target[row] = dot(updated_row, x[b,:])           (block reduction)
//   - g[row] = clamp(target - x[(b-1)%4, qw], -1, 1) / 9216
//   - new_kernel row = kernel row, minus LR*g*x[b,q,Q]*x[b,Q,w]/sqrt(2) at W==w
// kernel/space/updated/new_kernel are streamed once with non-temporal hints;
// bias (21 MB, reused 4x across b) stays on the regular cached path (L2=192MB).
// ---------------------------------------------------------------------------
__global__ __launch_bounds__(192)
void fused_fwd(const float* __restrict__ x,      // (4,48,48)
               const float* __restrict__ kern,   // (9216,2304)
               const float* __restrict__ bias,   // (2304,2304)
               const float* __restrict__ space,  // (9216,2304)
               float* __restrict__ target,       // (9216)
               float* __restrict__ updated,      // (9216,2304)
               float* __restrict__ nkern,        // (9216,2304)
               float* __restrict__ g_ws) {       // (9216) == G[b][qw]
  __shared__ float warp_part[6];
  __shared__ float sg;
  const int tid = threadIdx.x;
  const int row = blockIdx.x;
  const int b  = row / ROW;
  const int qw = row - b * ROW;
  const int q  = qw / DD;
  const int w  = qw - q * DD;

  const float* __restrict__ xb = x + (size_t)b * ROW;
  const v4f* __restrict__ k4 = (const v4f*)(kern  + (size_t)row * ROW);
  const v4f* __restrict__ s4 = (const v4f*)(space + (size_t)row * ROW);
  const v4f* __restrict__ b4 = (const v4f*)(bias  + (size_t)qw  * ROW);
  const v4f* __restrict__ x4 = (const v4f*)xb;
  v4f* __restrict__ u4 = (v4f*)(updated + (size_t)row * ROW);
  v4f* __restrict__ o4 = (v4f*)(nkern   + (size_t)row * ROW);

  v4f kv[3];                      // keep the kernel row slice in VGPRs
  float acc = 0.f;
  #pragma unroll
  for (int ch = 0; ch < 3; ++ch) {
    const int i    = tid + ch * 192;      // 0..575 (v4f index; 48%4==0 so no Q crossing)
    const int base = i * 4;
    const int Q    = base / DD;
    const int wb   = base - Q * DD;
    kv[ch] = __builtin_nontemporal_load(&k4[i]);
    const v4f sv = __builtin_nontemporal_load(&s4[i]);
    const v4f bv = b4[i];
    const v4f xv = x4[i];
    const float kr = xb[q * DD + Q] * INV_SQRT2;   // kron value when W==w
    v4f uv;
    uv.x = sv.x + bv.x + ((wb + 0 == w) ? kv[ch].x * kr : 0.f);
    uv.y = sv.y + bv.y + ((wb + 1 == w) ? kv[ch].y * kr : 0.f);
    uv.z = sv.z + bv.z + ((wb + 2 == w) ? kv[ch].z * kr : 0.f);
    uv.w = sv.w + bv.w + ((wb + 3 == w) ? kv[ch].w * kr : 0.f);
    acc += uv.x * xv.x + uv.y * xv.y + uv.z * xv.z + uv.w * xv.w;
    __builtin_nontemporal_store(uv, &u4[i]);
  }

  // wave32 reduction, then cross-wave via LDS
  #pragma unroll
  for (int off = 16; off > 0; off >>= 1) acc += __shfl_down(acc, off, 32);
  const int lane = tid & 31, wid = tid >> 5;
  if (lane == 0) warp_part[wid] = acc;
  __syncthreads();
  if (tid == 0) {
    float t = 0.f;
    #pragma unroll
    for (int i = 0; i < 6; ++i) t += warp_part[i];
    target[row] = t;
    const int   bprev = (b + 3) & 3;                      // roll(shift=1, axis=0)
    const float diff  = t - x[(size_t)bprev * ROW + qw];
    const float gg    = fminf(1.f, fmaxf(-1.f, diff)) * INV_N;  // huber' / N
    g_ws[row] = gg;
    sg = gg;
  }
  __syncthreads();

  // new_kernel: copy + sparse rank-1 correction at W==w (no second DRAM read of kern)
  const float c0 = LRF * sg * INV_SQRT2;
  #pragma unroll
  for (int ch = 0; ch < 3; ++ch) {
    const int i    = tid + ch * 192;
    const int base = i * 4;
    const int Q    = base / DD;
    const int wb   = base - Q * DD;
    v4f ov = kv[ch];
    const int cidx = w - wb;
    if (cidx >= 0 && cidx < 4) {
      const float delta = c0 * xb[q * DD + Q] * xb[Q * DD + w];
      ov[cidx] -= delta;
    }
    __builtin_nontemporal_store(ov, &o4[i]);
  }
}

// ---------------------------------------------------------------------------
// Kernel 2: new_bias = bias - LR * (G^T X), a (2304x4)*(4x2304) GEMM (K=4).
// Exact fit for V_WMMA_F32_16X16X4_F32: one wave produces one 16x16 tile.
// A-layout (ISA 16x4 f32): lanes 0-15 hold K=0(v0),K=1(v1); lanes 16-31 K=2,3.
// B assumed symmetric (lanes 0-15: K=0,1; lanes 16-31: K=2,3 for cols N=0..15).
// C/D layout: VGPR r -> row r (lanes 0-15) / row r+8 (lanes 16-31).
// ---------------------------------------------------------------------------
#define NTILE 144  // 2304 / 16

__global__ __launch_bounds__(256)
void bias_update_wmma(const float* __restrict__ bias,   // (2304,2304)
                      const float* __restrict__ g,      // (4,2304)  G[k][i]
                      const float* __restrict__ x,      // (4,2304)  X[k][j]
                      float* __restrict__ nbias) {      // (2304,2304)
  const int wave = (blockIdx.x * 256 + threadIdx.x) >> 5;  // grid sized exactly
  const int lane = threadIdx.x & 31;
  const int ti = wave / NTILE;
  const int tj = wave - ti * NTILE;
  const int i0 = ti * 16, j0 = tj * 16;
  const int half = lane >> 4;       // 0: K=0,1   1: K=2,3
  const int m    = lane & 15;
  const int kb   = half * 2;

  v2f a, bm;
  a.x  = g[(size_t)(kb + 0) * ROW + i0 + m];   // A[m][k] = G[k][i0+m]
  a.y  = g[(size_t)(kb + 1) * ROW + i0 + m];
  bm.x = x[(size_t)(kb + 0) * ROW + j0 + m];   // B[k][n] = X[k][j0+n]
  bm.y = x[(size_t)(kb + 1) * ROW + j0 + m];

  v8f d;
#if __has_builtin(__builtin_amdgcn_wmma_f32_16x16x4_f32)
  v8f c = {};
  d = __builtin_amdgcn_wmma_f32_16x16x4_f32(
      /*neg_a=*/false, a, /*neg_b=*/false, bm,
      /*c_mod=*/(short)0, c, /*reuse_a=*/false, /*reuse_b=*/false);
#else
  // Correct scalar fallback (compile-safety only; histogram shows if taken)
  #pragma unroll
  for (int r = 0; r < 8; ++r) {
    const int rowi = i0 + r + half * 8;
    float s = 0.f;
    #pragma unroll
    for (int k = 0; k < 4; ++k)
      s += g[(size_t)k * ROW + rowi] * x[(size_t)k * ROW + j0 + m];
    d[r] = s;
  }
#endif

  #pragma unroll
  for (int r = 0; r < 8; ++r) {
    const int rowi = i0 + r + half * 8;          // D VGPR r -> row r / r+8
    const size_t idx = (size_t)rowi * ROW + j0 + m;
    const float nb = bias[idx] - LRF * d[r];
    __builtin_nontemporal_store(nb, &nbias[idx]);
  }
}

// ---------------------------------------------------------------------------
extern "C" void kernel_launch(void* const* d_in, const int* in_sizes, int n_in,
                              void* d_out, int out_size, void* d_ws, size_t ws_size,
                              hipStream_t stream) {
  const float* x     = (const float*)d_in[0];  // (4,48,48)
  const float* kern  = (const float*)d_in[1];  // (4,48,48,48,48)
  const float* bias  = (const float*)d_in[2];  // (48,48,48,48)
  const float* space = (const float*)d_in[3];  // (4,48,48,48,48)

  float* out     = (float*)d_out;              // tuple, flat in return order:
  float* target  = out;                                    //  9216
  float* nkern   = out + 9216;                             //  21233664
  float* nbias   = out + 9216 + 21233664;                  //  5308416
  float* updated = out + 9216 + 21233664 + 5308416;        //  21233664
  float* g_ws    = (float*)d_ws;                           //  9216 floats

  fused_fwd<<<NROWS, 192, 0, stream>>>(x, kern, bias, space,
                                       target, updated, nkern, g_ws);
  // 144*144 = 20736 tiles, 8 waves/block -> 2592 blocks exactly
  bias_update_wmma<<<(NTILE * NTILE) / 8, 256, 0, stream>>>(bias, g_ws, x, nbias);
}